// MultiResolutionHashEncoding_7902739825031
// MI455X (gfx1250) — compile-verified
//
#include <hip/hip_runtime.h>
#include <cmath>
#include <cstdint>

#define NLEV 16
#define BLOCK 256
#define STRIDE_W 34  // LDS words per point: 32 feats + 2 pad (conflict-free b64, keeps 8B align)

struct HashParams {
  float    resf[NLEV];  // level resolution as f32 (exact, <= 2048)
  unsigned d[NLEV];     // table size
  unsigned m[NLEV];     // magic multiplier: floor(2^(31+ell)/d)+1
  unsigned sh[NLEV];    // ell-1 (post-mulhi shift)
};

__global__ __launch_bounds__(BLOCK) void hash_encode_kernel(
    const float* __restrict__ coords, const float* __restrict__ tables,
    float* __restrict__ out, HashParams P) {
  __shared__ float sbuf[BLOCK * STRIDE_W];

  const unsigned t = threadIdx.x;
  const unsigned long long pt = (unsigned long long)blockIdx.x * BLOCK + t;
  // Coordinates are streamed exactly once: non-temporal, don't pollute L2
  const float* cptr = coords + pt * 3;
  const float cx = __builtin_nontemporal_load(cptr + 0);
  const float cy = __builtin_nontemporal_load(cptr + 1);
  const float cz = __builtin_nontemporal_load(cptr + 2);
  const float2* __restrict__ tab2 = (const float2*)tables;

#pragma unroll
  for (int l = 0; l < NLEV; ++l) {
    const float rf = P.resf[l];
    // coords in [0,1): trunc == floor; identical f32 RNE multiply to reference
    int gx = (int)(cx * rf);
    int gy = (int)(cy * rf);
    int gz = (int)(cz * rf);
    // int32 wraparound hash (matches TF/JAX int32 semantics)
    int h = (int)((unsigned)gx * 73856093u ^
                  (unsigned)gy * 19349663u ^
                  (unsigned)gz * 83492791u);
    unsigned u = (h < 0) ? (0u - (unsigned)h) : (unsigned)h;  // |h| with wrap; u in [0, 2^31]
    // floored mod via magic division: exact for all u <= 2^31 with our divisors
    unsigned q = (unsigned)(((unsigned long long)u * P.m[l]) >> 32) >> P.sh[l];
    unsigned r = u - q * P.d[l];
    // jnp: abs(INT_MIN) wraps to INT_MIN, then floored % -> (d - r) % d
    if ((h == (int)0x80000000) && (r != 0u)) r = P.d[l] - r;

    float2 f = tab2[((size_t)l << 19) + r];  // stacked tables (RT caching: hot in L2)
    sbuf[t * STRIDE_W + 2 * l + 0] = f.x;
    sbuf[t * STRIDE_W + 2 * l + 1] = f.y;
  }

  __syncthreads();

  // Coalesced writeback: LDS -> global via CDNA5 async DMA (ASYNCcnt path).
  // Block chunk = 256 points * 32 f32 = 32 KB contiguous; 16 iterations of
  // fully-coalesced b64 per lane. Output is write-once: non-temporal so the
  // 256 MB stream does not evict the L2-resident hash tables.
  const unsigned ldsbase = (unsigned)(uintptr_t)(&sbuf[0]);     // LDS byte offset (low 32 bits)
  const unsigned chunk = blockIdx.x * (BLOCK * NLEV * 2u * 4u); // byte offset of block's region
  unsigned lofs = ldsbase + (((t >> 4) * STRIDE_W) + (t & 15u) * 2u) * 4u;  // 8B-aligned
  unsigned gofs = chunk + t * 8u;                                           // coalesced b64
#pragma unroll
  for (int k = 0; k < NLEV; ++k) {
    asm volatile("global_store_async_from_lds_b64 %0, %1, %2 th:TH_STORE_NT"
                 :: "v"(gofs), "v"(lofs), "s"(out)
                 : "memory");
    lofs += 16u * STRIDE_W * 4u;  // next 16 points' rows in LDS (+2176 B)
    gofs += BLOCK * 8u;           // next coalesced 2 KB slab (+2048 B)
  }
  asm volatile("s_wait_asynccnt 0x0" ::: "memory");
}

extern "C" void kernel_launch(void* const* d_in, const int* in_sizes, int n_in,
                              void* d_out, int out_size, void* d_ws, size_t ws_size,
                              hipStream_t stream) {
  const float* coords = (const float*)d_in[0];
  const float* tables = (const float*)d_in[1];
  float* out = (float*)d_out;
  (void)d_ws; (void)ws_size; (void)n_in; (void)out_size;

  // Replicate numpy's float64 level math exactly (glibc libm, same as numpy),
  // and build magic-division constants so the GPU never divides.
  HashParams P;
  const double growth = std::exp((std::log(2048.0) - std::log(16.0)) / 15.0);
  for (int l = 0; l < NLEV; ++l) {
    long long res  = (long long)std::floor(16.0 * std::pow(growth, (double)l));
    long long cube = res * res * res;
    long long cap  = 1ll << 19;
    unsigned  d    = (unsigned)(cube < cap ? cube : cap);
    unsigned ell = 0;
    while ((1ull << ell) < (unsigned long long)d) ++ell;   // ceil(log2(d)), >= 12 here
    unsigned long long m = ((1ull << (31 + ell)) / d) + 1ull;  // fits in 32 bits for all levels
    P.resf[l] = (float)res;
    P.d[l]    = d;
    P.m[l]    = (unsigned)m;
    P.sh[l]   = ell - 1;
  }

  const int n_points = in_sizes[0] / 3;      // 2^21
  const int blocks   = n_points / BLOCK;     // exactly divisible
  hash_encode_kernel<<<blocks, BLOCK, 0, stream>>>(coords, tables, out, P);
}